// GINEModel_78374563217910
// MI455X (gfx1250) — compile-verified
//
#include <hip/hip_runtime.h>
#include <hip/hip_bf16.h>
#include <math.h>

// GINE model on gfx1250 (MI455X, wave32):
//  - fp32 WMMA (V_WMMA_F32_16X16X4_F32) for all dense GEMMs
//  - LDS-staged operands; weights stored k-pair interleaved so every WMMA
//    B-fragment is a single conflict-free ds_load_b64 into an even VGPR pair
//  - fused edge kernel: edge GEMM + h[src] gather + relu + f32 atomic scatter
//  - BatchNorm via per-column atomic stats, f32 atomic pooling, scalar head
// Working set (~65 MB) fits in the 192 MB L2 -> kernel is L2/LDS bound.

typedef __attribute__((ext_vector_type(2))) float v2f;
typedef __attribute__((ext_vector_type(8))) float v8f;

#define NN 50000
#define EE 800000
#define DD 64
#define GG 512
#define FF 10

// k-pair interleaved weight pitch (floats per k-pair row of 64 cols):
// lane n reads banks {2n,2n+1}; +160 floats == +32 banks -> lanes 16..31
// land on banks 32..63: conflict-free wave32 b64 reads.
#define WPITCH 160

// ---------------------------------------------------------------------------
__global__ void zero4_kernel(float4* __restrict__ p, int n4) {
    int i = blockIdx.x * blockDim.x + threadIdx.x;
    if (i < n4) p[i] = make_float4(0.f, 0.f, 0.f, 0.f);
}

// ---------------------------------------------------------------------------
// Fused edge kernel. Per wave: one tile of 16 edges.
//   msg = relu(h[src] + edge_attr @ We + be);  agg[dst] += msg
#define EPITCH 20   // A-tile pitch: 20m mod 64 distinct for m=0..15; float4 aligned
__global__ __launch_bounds__(256) void edge_msg_kernel(
    const float* __restrict__ h, const float* __restrict__ edge_attr,
    const float* __restrict__ We, const float* __restrict__ be,
    const int* __restrict__ src, const int* __restrict__ dst,
    float* __restrict__ agg, int nTiles) {
    __shared__ __align__(16) float ldsWe[8 * WPITCH];       // 16x64 k-pair interleaved, 5 KB
    __shared__ __align__(16) float ldsE[8][16 * EPITCH];    // 10 KB

    int tid  = threadIdx.x;
    int w    = tid >> 5;
    int lane = tid & 31;
    int tile = blockIdx.x * 8 + w;
    int teff = tile < nTiles ? tile : nTiles - 1;   // clamp: all waves stage+compute

    // stage We (16x64) k-pair interleaved: element (k,n) -> (k>>1)*WPITCH + 2n + (k&1)
    {
        float4 v = ((const float4*)We)[tid];        // k = tid>>4, cols 4*(tid&15)..+3
        int k = tid >> 4;
        int c = (tid & 15) * 4;
        float* p = &ldsWe[(k >> 1) * WPITCH + (k & 1)];
        p[(c + 0) * 2] = v.x;
        p[(c + 1) * 2] = v.y;
        p[(c + 2) * 2] = v.z;
        p[(c + 3) * 2] = v.w;
    }
    // stage this wave's 16x16 edge_attr tile: 2 coalesced float4 per lane
    {
        const float4* ge = (const float4*)(edge_attr + (size_t)teff * 256 + lane * 8);
        int r  = lane >> 1;
        int cb = (lane & 1) * 8;
        float4 e0 = ge[0], e1 = ge[1];
        *(float4*)&ldsE[w][r * EPITCH + cb]     = e0;
        *(float4*)&ldsE[w][r * EPITCH + cb + 4] = e1;
    }
    __syncthreads();

    int ml   = lane & 15;
    int khalf = lane >> 4;          // 0: k-pair (ks,ks+1), 1: (ks+2,ks+3)
    int n0   = lane & 15;

    v8f acc0 = {}, acc1 = {}, acc2 = {}, acc3 = {};
    for (int ks = 0; ks < 16; ks += 4) {
        v2f a = *(const v2f*)&ldsE[w][ml * EPITCH + ks + khalf * 2];
        const float* wrow = &ldsWe[((ks >> 1) + khalf) * WPITCH];
        v2f b0 = *(const v2f*)&wrow[(0  + n0) * 2];
        v2f b1 = *(const v2f*)&wrow[(16 + n0) * 2];
        v2f b2 = *(const v2f*)&wrow[(32 + n0) * 2];
        v2f b3 = *(const v2f*)&wrow[(48 + n0) * 2];
        acc0 = __builtin_amdgcn_wmma_f32_16x16x4_f32(false, a, false, b0, (short)0, acc0, false, false);
        acc1 = __builtin_amdgcn_wmma_f32_16x16x4_f32(false, a, false, b1, (short)0, acc1, false, false);
        acc2 = __builtin_amdgcn_wmma_f32_16x16x4_f32(false, a, false, b2, (short)0, acc2, false, false);
        acc3 = __builtin_amdgcn_wmma_f32_16x16x4_f32(false, a, false, b3, (short)0, acc3, false, false);
    }

    if (tile >= nTiles) return;   // wave-uniform; no barriers after this point

    int mhi = (lane >> 4) * 8;    // C layout: lanes<16 -> rows 0..7, lanes>=16 -> rows 8..15
    v8f accs[4] = {acc0, acc1, acc2, acc3};
    for (int t = 0; t < 4; ++t) {
        int col  = t * 16 + n0;
        float bv = be[col];
        for (int v = 0; v < 8; ++v) {
            int e = tile * 16 + mhi + v;
            float val = accs[t][v] + bv;
            int s = src[e];
            val += h[(size_t)s * DD + col];     // coalesced 64B half-wave gather
            val = fmaxf(val, 0.0f);
            int d = dst[e];
            atomicAdd(&agg[(size_t)d * DD + col], val);
        }
    }
}

// ---------------------------------------------------------------------------
// Specialized node GEMM (K=64, N=64): out = act((A [+ A2]) @ W + bias)
// W (64x64) staged k-pair interleaved per block (20 KB); each wave stages its
// 16x64 A tile (fused A+A2 add). 64 WMMAs per wave, 4 accumulators.
#define APITCH 68   // 68m mod 64 = 4m: distinct banks for m=0..15; float4 aligned
__global__ __launch_bounds__(256) void gemm64_kernel(
    const float* __restrict__ A, const float* __restrict__ A2,
    const float* __restrict__ W, const float* __restrict__ bias,
    float* __restrict__ out, int tilesM, int relu_flag,
    float* __restrict__ colsum, float* __restrict__ colsumsq) {
    __shared__ __align__(16) float ldsW[32 * WPITCH];       // 64x64 k-pair interleaved, 20 KB
    __shared__ __align__(16) float ldsA[8][16 * APITCH];    // 34 KB

    int tid  = threadIdx.x;
    int w    = tid >> 5;
    int lane = tid & 31;
    int mt   = blockIdx.x * 8 + w;
    int meff = mt < tilesM ? mt : tilesM - 1;

    // stage W (64x64) k-pair interleaved: 16 floats (4 float4) per thread
    {
        const float4* gw = (const float4*)W;
        for (int t = 0; t < 4; ++t) {
            int i4 = t * 256 + tid;             // float4 index; 16 per 64-float row
            float4 v = gw[i4];
            int k = i4 >> 4;
            int c = (i4 & 15) * 4;
            float* p = &ldsW[(k >> 1) * WPITCH + (k & 1)];
            p[(c + 0) * 2] = v.x;
            p[(c + 1) * 2] = v.y;
            p[(c + 2) * 2] = v.z;
            p[(c + 3) * 2] = v.w;
        }
    }
    // stage this wave's 16x64 A tile (+A2), coalesced: 8 float4 per lane
    {
        int r  = lane >> 1;
        int cb = (lane & 1) * 32;
        const float4* ga  = (const float4*)(A + (size_t)(meff * 16 + r) * DD + cb);
        const float4* ga2 = A2 ? (const float4*)(A2 + (size_t)(meff * 16 + r) * DD + cb)
                               : (const float4*)0;
        for (int j = 0; j < 8; ++j) {
            float4 v = ga[j];
            if (ga2) {
                float4 v2 = ga2[j];
                v.x += v2.x; v.y += v2.y; v.z += v2.z; v.w += v2.w;
            }
            *(float4*)&ldsA[w][r * APITCH + cb + j * 4] = v;
        }
    }
    __syncthreads();

    int ml    = lane & 15;
    int khalf = lane >> 4;
    int n0    = lane & 15;

    v8f acc0 = {}, acc1 = {}, acc2 = {}, acc3 = {};
    for (int ks = 0; ks < DD; ks += 4) {
        v2f a = *(const v2f*)&ldsA[w][ml * APITCH + ks + khalf * 2];
        const float* wrow = &ldsW[((ks >> 1) + khalf) * WPITCH];
        v2f b0 = *(const v2f*)&wrow[(0  + n0) * 2];
        v2f b1 = *(const v2f*)&wrow[(16 + n0) * 2];
        v2f b2 = *(const v2f*)&wrow[(32 + n0) * 2];
        v2f b3 = *(const v2f*)&wrow[(48 + n0) * 2];
        acc0 = __builtin_amdgcn_wmma_f32_16x16x4_f32(false, a, false, b0, (short)0, acc0, false, false);
        acc1 = __builtin_amdgcn_wmma_f32_16x16x4_f32(false, a, false, b1, (short)0, acc1, false, false);
        acc2 = __builtin_amdgcn_wmma_f32_16x16x4_f32(false, a, false, b2, (short)0, acc2, false, false);
        acc3 = __builtin_amdgcn_wmma_f32_16x16x4_f32(false, a, false, b3, (short)0, acc3, false, false);
    }

    if (mt >= tilesM) return;     // wave-uniform; after all barriers

    int mhi = (lane >> 4) * 8;
    v8f accs[4] = {acc0, acc1, acc2, acc3};
    for (int t = 0; t < 4; ++t) {
        int col  = t * 16 + n0;
        float bv = bias[col];
        float s = 0.f, sq = 0.f;
        for (int v = 0; v < 8; ++v) {
            float val = accs[t][v] + bv;
            if (relu_flag) val = fmaxf(val, 0.f);
            out[(size_t)(mt * 16 + mhi + v) * DD + col] = val;
            s  += val;
            sq += val * val;
        }
        if (colsum) {
            atomicAdd(&colsum[col], s);
            atomicAdd(&colsumsq[col], sq);
        }
    }
}

// ---------------------------------------------------------------------------
// Generic WMMA f32 GEMM (used for the small 512x192x192 lin1).
__global__ __launch_bounds__(256) void gemm_wmma_kernel(
    const float* __restrict__ A, const float* __restrict__ W,
    const float* __restrict__ bias, float* __restrict__ out,
    int M, int K, int Ncols, int relu_flag) {
    int wave = blockIdx.x * (blockDim.x >> 5) + (threadIdx.x >> 5);
    int lane = threadIdx.x & 31;
    int tiles_n = Ncols >> 4;
    int tiles_m = M >> 4;
    if (wave >= tiles_m * tiles_n) return;
    int tm = wave / tiles_n;
    int tn = wave - tm * tiles_n;

    int ml   = lane & 15;
    int koff = (lane >> 4) * 2;
    const float* arow = A + (size_t)(tm * 16 + ml) * K;
    int n0 = tn * 16 + (lane & 15);

    v8f acc = {};
    for (int ks = 0; ks < K; ks += 4) {
        v2f a;
        a.x = arow[ks + koff];
        a.y = arow[ks + koff + 1];
        v2f b;
        b.x = W[(size_t)(ks + koff) * Ncols + n0];
        b.y = W[(size_t)(ks + koff + 1) * Ncols + n0];
        acc = __builtin_amdgcn_wmma_f32_16x16x4_f32(false, a, false, b, (short)0, acc, false, false);
    }

    int mhi = (lane >> 4) * 8;
    float bv = bias[n0];
    for (int v = 0; v < 8; ++v) {
        float val = acc[v] + bv;
        if (relu_flag) val = fmaxf(val, 0.f);
        out[(size_t)(tm * 16 + mhi + v) * Ncols + n0] = val;
    }
}

// ---------------------------------------------------------------------------
__global__ void bn_relu_kernel(float* __restrict__ z,
                               const float* __restrict__ colsum,
                               const float* __restrict__ colsumsq,
                               const float* __restrict__ g, const float* __restrict__ b,
                               int total, float invN) {
    int i = blockIdx.x * blockDim.x + threadIdx.x;
    if (i >= total) return;
    int col = i & (DD - 1);
    float mu  = colsum[col] * invN;
    float var = colsumsq[col] * invN - mu * mu;
    float rstd = rsqrtf(var + 1e-5f);
    float val = (z[i] - mu) * rstd * g[col] + b[col];
    z[i] = fmaxf(val, 0.0f);
}

// ---------------------------------------------------------------------------
__global__ void count_kernel(const int* __restrict__ batch, float* __restrict__ counts, int n) {
    int i = blockIdx.x * blockDim.x + threadIdx.x;
    if (i < n) atomicAdd(&counts[batch[i]], 1.0f);
}

__global__ void pool_kernel(const int* __restrict__ batch,
                            const float* __restrict__ H1, const float* __restrict__ H2,
                            const float* __restrict__ H3, float* __restrict__ pooled, int n) {
    int idx = blockIdx.x * blockDim.x + threadIdx.x;
    if (idx >= n * DD) return;
    int node = idx >> 6;
    int d = idx & (DD - 1);
    int g = batch[node];
    atomicAdd(&pooled[(size_t)g * 192 + d],       H1[idx]);
    atomicAdd(&pooled[(size_t)g * 192 + 64 + d],  H2[idx]);
    atomicAdd(&pooled[(size_t)g * 192 + 128 + d], H3[idx]);
}

__global__ void pool_div_kernel(float* __restrict__ pooled, const float* __restrict__ counts, int total) {
    int idx = blockIdx.x * blockDim.x + threadIdx.x;
    if (idx >= total) return;
    float c = fmaxf(counts[idx / 192], 1.0f);
    pooled[idx] /= c;
}

// ---------------------------------------------------------------------------
__global__ void head_kernel(const float* __restrict__ hc, const float* __restrict__ W,
                            const float* __restrict__ b, float* __restrict__ out, int g_) {
    int g = blockIdx.x * blockDim.x + threadIdx.x;
    if (g >= g_) return;
    float logit[FF];
    for (int f = 0; f < FF; ++f) {
        float s = b[f];
        for (int k = 0; k < 192; ++k) s += hc[(size_t)g * 192 + k] * W[k * FF + f];
        logit[f] = s;
    }
    float m = logit[0];
    for (int f = 1; f < FF; ++f) m = fmaxf(m, logit[f]);
    float se = 0.0f;
    for (int f = 0; f < FF; ++f) se += expf(logit[f] - m);
    float ls = logf(se);
    for (int f = 0; f < FF; ++f) out[(size_t)g * FF + f] = logit[f] - m - ls;
}

// ---------------------------------------------------------------------------
extern "C" void kernel_launch(void* const* d_in, const int* in_sizes, int n_in,
                              void* d_out, int out_size, void* d_ws, size_t ws_size,
                              hipStream_t stream) {
    (void)in_sizes; (void)n_in; (void)out_size; (void)ws_size;
    const float* x         = (const float*)d_in[0];
    const int*   edge_idx  = (const int*)d_in[1];
    const float* edge_attr = (const float*)d_in[2];
    const int*   batch     = (const int*)d_in[3];
    const float* We   = (const float*)d_in[4];
    const float* be   = (const float*)d_in[5];
    const float* W1   = (const float*)d_in[6];
    const float* b1   = (const float*)d_in[7];
    const float* bn_g = (const float*)d_in[8];
    const float* bn_b = (const float*)d_in[9];
    const float* W2   = (const float*)d_in[10];
    const float* b2   = (const float*)d_in[11];
    const float* l1W  = (const float*)d_in[12];
    const float* l1b  = (const float*)d_in[13];
    const float* l2W  = (const float*)d_in[14];
    const float* l2b  = (const float*)d_in[15];
    float* out = (float*)d_out;

    const int* src = edge_idx;
    const int* dst = edge_idx + EE;

    // workspace layout (floats)
    float* ws = (float*)d_ws;
    float* AGG    = ws;                              // N*64
    float* Z      = AGG + (size_t)NN * DD;           // N*64
    float* H1     = Z + (size_t)NN * DD;             // N*64
    float* H2     = H1 + (size_t)NN * DD;            // N*64
    float* H3     = H2 + (size_t)NN * DD;            // N*64
    float* STAT   = H3 + (size_t)NN * DD;            // 128 (colsum | colsumsq)
    float* POOLED = STAT + 128;                      // G*192
    float* COUNTS = POOLED + (size_t)GG * 192;       // G
    float* HC     = COUNTS + GG;                     // G*192

    const float* hin = x;
    float* houts[3] = {H1, H2, H3};

    int tilesM = NN / 16;     // 3125
    int tilesE = EE / 16;     // 50000

    for (int L = 0; L < 3; ++L) {
        zero4_kernel<<<(NN * DD / 4 + 255) / 256, 256, 0, stream>>>((float4*)AGG, NN * DD / 4);
        zero4_kernel<<<1, 32, 0, stream>>>((float4*)STAT, 32);

        edge_msg_kernel<<<(tilesE + 7) / 8, 256, 0, stream>>>(
            hin, edge_attr, We + (size_t)L * 16 * DD, be + (size_t)L * DD,
            src, dst, AGG, tilesE);

        // z = (h + agg) @ W1 + b1, accumulate BN stats
        gemm64_kernel<<<(tilesM + 7) / 8, 256, 0, stream>>>(
            hin, AGG, W1 + (size_t)L * DD * DD, b1 + (size_t)L * DD,
            Z, tilesM, 0, STAT, STAT + 64);

        bn_relu_kernel<<<(NN * DD + 255) / 256, 256, 0, stream>>>(
            Z, STAT, STAT + 64, bn_g + (size_t)L * DD, bn_b + (size_t)L * DD,
            NN * DD, 1.0f / (float)NN);

        // h = relu(z @ W2 + b2)
        gemm64_kernel<<<(tilesM + 7) / 8, 256, 0, stream>>>(
            Z, (const float*)0, W2 + (size_t)L * DD * DD, b2 + (size_t)L * DD,
            houts[L], tilesM, 1, (float*)0, (float*)0);

        hin = houts[L];
    }

    // global mean pool per graph
    zero4_kernel<<<(GG * 192 / 4 + 255) / 256, 256, 0, stream>>>((float4*)POOLED, GG * 192 / 4);
    zero4_kernel<<<1, 128, 0, stream>>>((float4*)COUNTS, GG / 4);
    count_kernel<<<(NN + 255) / 256, 256, 0, stream>>>(batch, COUNTS, NN);
    pool_kernel<<<(NN * DD + 255) / 256, 256, 0, stream>>>(batch, H1, H2, H3, POOLED, NN);
    pool_div_kernel<<<(GG * 192 + 255) / 256, 256, 0, stream>>>(POOLED, COUNTS, GG * 192);

    // hc = relu(pooled @ lin1_W + lin1_b)   (512 x 192 x 192)
    int mlpTiles = (GG / 16) * (192 / 16);
    gemm_wmma_kernel<<<(mlpTiles + 7) / 8, 256, 0, stream>>>(
        POOLED, l1W, l1b, HC, GG, 192, 192, 1);

    // logits + log_softmax
    head_kernel<<<(GG + 255) / 256, 256, 0, stream>>>(HC, l2W, l2b, out, GG);
}